// GINGraph_30932354466092
// MI455X (gfx1250) — compile-verified
//
#include <hip/hip_runtime.h>
#include <hip/hip_bf16.h>

// ---------------------------------------------------------------------------
// GIN graph network for MI455X (gfx1250), fp32 path with f32 WMMA GEMMs.
// ---------------------------------------------------------------------------

#define N_NODES   100000
#define N_EDGES   1600000
#define DIM       64
#define DIM2      128
#define N_LAYERS  4
#define N_GRAPHS  1024

typedef __attribute__((ext_vector_type(2))) float v2f;
typedef __attribute__((ext_vector_type(8))) float v8f;

__device__ __forceinline__ v8f wmma_f32_k4(v2f a, v2f b, v8f c) {
    // D = A(16x4 f32) * B(4x16 f32) + C(16x16 f32)
    return __builtin_amdgcn_wmma_f32_16x16x4_f32(
        /*neg_a=*/false, a, /*neg_b=*/false, b,
        /*c_mod=*/(short)0, c, /*reuse_a=*/false, /*reuse_b=*/false);
}

// ---------------------------------------------------------------------------
// Utility: zero a float buffer
// ---------------------------------------------------------------------------
__global__ void zero_f(float* __restrict__ p, int n) {
    int i = blockIdx.x * blockDim.x + threadIdx.x;
    if (i < n) p[i] = 0.0f;
}

// ---------------------------------------------------------------------------
// Atom encoder: h[n,:] = sum_{c=0..8} atom_emb[c, x[n,c], :]
// 16 threads per node, float4 per thread.
// ---------------------------------------------------------------------------
__global__ void atom_encode(const int* __restrict__ x,
                            const float* __restrict__ aemb,   // [9,100,64]
                            float* __restrict__ h) {
    int gid = blockIdx.x * blockDim.x + threadIdx.x;
    if (gid >= N_NODES * 16) return;
    int n = gid >> 4;
    int d = (gid & 15) * 4;
    float4 acc = make_float4(0.f, 0.f, 0.f, 0.f);
#pragma unroll
    for (int c = 0; c < 9; ++c) {
        int v = x[n * 9 + c];
        const float4 t = *(const float4*)(aemb + (size_t)(c * 100 + v) * DIM + d);
        acc.x += t.x; acc.y += t.y; acc.z += t.z; acc.w += t.w;
    }
    *(float4*)(h + (size_t)n * DIM + d) = acc;
}

// ---------------------------------------------------------------------------
// Edge phase: ef = bond_embed(edge_attr); msg = relu(h[dst] + ef);
//             agg[src] += msg  (f32 global atomics -> L2 atomic units)
// 16 threads per edge, float4 per thread.
// ---------------------------------------------------------------------------
__global__ void edge_scatter(const int* __restrict__ edge_index, // [2,E]
                             const int* __restrict__ edge_attr,  // [E,3]
                             const float* __restrict__ bemb,     // [3,8,64] (layer slice)
                             const float* __restrict__ h,
                             float* __restrict__ agg) {
    int gid = blockIdx.x * blockDim.x + threadIdx.x;
    if (gid >= N_EDGES * 16) return;
    int e = gid >> 4;
    int d = (gid & 15) * 4;
    int src = edge_index[e];
    int dst = edge_index[N_EDGES + e];
    int a0 = edge_attr[e * 3 + 0];
    int a1 = edge_attr[e * 3 + 1];
    int a2 = edge_attr[e * 3 + 2];
    float4 t0 = *(const float4*)(bemb + (size_t)(0 * 8 + a0) * DIM + d);
    float4 t1 = *(const float4*)(bemb + (size_t)(1 * 8 + a1) * DIM + d);
    float4 t2 = *(const float4*)(bemb + (size_t)(2 * 8 + a2) * DIM + d);
    float4 hd = *(const float4*)(h + (size_t)dst * DIM + d);
    float m0 = fmaxf(hd.x + t0.x + t1.x + t2.x, 0.f);
    float m1 = fmaxf(hd.y + t0.y + t1.y + t2.y, 0.f);
    float m2 = fmaxf(hd.z + t0.z + t1.z + t2.z, 0.f);
    float m3 = fmaxf(hd.w + t0.w + t1.w + t2.w, 0.f);
    float* dsta = agg + (size_t)src * DIM + d;
    atomicAdd(dsta + 0, m0);
    atomicAdd(dsta + 1, m1);
    atomicAdd(dsta + 2, m2);
    atomicAdd(dsta + 3, m3);
}

// ---------------------------------------------------------------------------
// GEMM1: Z1[N,128] = ((1+eps)*h + agg)[N,64] @ W1[64,128] + b1
// One wave32 per 16x16 tile; V_WMMA_F32_16X16X4_F32, K-loop of 16.
// A fragment: lanes 0-15 -> K = k0+0..1, lanes 16-31 -> K = k0+2..3 (row = lane&15)
// B fragment: same K split, col = lane&15.
// C VGPR v: lanes 0-15 -> row v, lanes 16-31 -> row v+8; col = lane&15.
// ---------------------------------------------------------------------------
__global__ void gemm1_wmma(const float* __restrict__ h,
                           const float* __restrict__ agg,
                           const float* __restrict__ eps_arr, int layer,
                           const float* __restrict__ W,    // [64,128]
                           const float* __restrict__ bias, // [128]
                           float* __restrict__ out) {      // [N,128]
    const int lane = threadIdx.x & 31;
    const int kh   = lane >> 4;          // 0 or 1
    const int lidx = lane & 15;
    const int row  = blockIdx.x * 16 + lidx;
    const int col  = blockIdx.y * 16 + lidx;
    const float ep = 1.0f + eps_arr[layer];
    const float* arow = h   + (size_t)row * DIM;
    const float* grow = agg + (size_t)row * DIM;
    v8f acc = {};
#pragma unroll
    for (int k0 = 0; k0 < DIM; k0 += 4) {
        const int k = k0 + kh * 2;
        v2f a, b;
        a.x = ep * arow[k]     + grow[k];
        a.y = ep * arow[k + 1] + grow[k + 1];
        b.x = W[(size_t)k * DIM2 + col];
        b.y = W[(size_t)(k + 1) * DIM2 + col];
        acc = wmma_f32_k4(a, b, acc);
    }
    const int r0 = blockIdx.x * 16 + kh * 8;
    const float bc = bias[col];
#pragma unroll
    for (int v = 0; v < 8; ++v)
        out[(size_t)(r0 + v) * DIM2 + col] = acc[v] + bc;
}

// ---------------------------------------------------------------------------
// GEMM2: H[N,64] = Z1n[N,128] @ W2[128,64] + b2   (written in place into h)
// ---------------------------------------------------------------------------
__global__ void gemm2_wmma(const float* __restrict__ A,    // [N,128]
                           const float* __restrict__ W,    // [128,64]
                           const float* __restrict__ bias, // [64]
                           float* __restrict__ out) {      // [N,64]
    const int lane = threadIdx.x & 31;
    const int kh   = lane >> 4;
    const int lidx = lane & 15;
    const int row  = blockIdx.x * 16 + lidx;
    const int col  = blockIdx.y * 16 + lidx;
    const float* arow = A + (size_t)row * DIM2;
    v8f acc = {};
#pragma unroll
    for (int k0 = 0; k0 < DIM2; k0 += 4) {
        const int k = k0 + kh * 2;
        v2f a, b;
        a.x = arow[k];
        a.y = arow[k + 1];
        b.x = W[(size_t)k * DIM + col];
        b.y = W[(size_t)(k + 1) * DIM + col];
        acc = wmma_f32_k4(a, b, acc);
    }
    const int r0 = blockIdx.x * 16 + kh * 8;
    const float bc = bias[col];
#pragma unroll
    for (int v = 0; v < 8; ++v)
        out[(size_t)(r0 + v) * DIM + col] = acc[v] + bc;
}

// ---------------------------------------------------------------------------
// Batchnorm stats: per-column sum and sum-of-squares (atomic reduction).
// blockDim.x must be a multiple of cols (256 with cols 64/128).
// sums[0..cols) = sum, sums[cols..2*cols) = sumsq
// ---------------------------------------------------------------------------
__global__ void bn_stats(const float* __restrict__ Z, int cols,
                         float* __restrict__ sums) {
    int col = threadIdx.x % cols;
    int rpb = blockDim.x / cols;
    int r = blockIdx.x * rpb + threadIdx.x / cols;
    int stride = gridDim.x * rpb;
    float s = 0.f, q = 0.f;
    for (; r < N_NODES; r += stride) {
        float v = Z[(size_t)r * cols + col];
        s += v;
        q += v * v;
    }
    atomicAdd(&sums[col], s);
    atomicAdd(&sums[cols + col], q);
}

// ---------------------------------------------------------------------------
// Batchnorm apply (+ optional ReLU), biased variance like the reference.
// ---------------------------------------------------------------------------
__global__ void bn_apply(const float* __restrict__ Z, float* __restrict__ out,
                         const float* __restrict__ sums,
                         const float* __restrict__ g, const float* __restrict__ b,
                         int cols, int do_relu) {
    int gid = blockIdx.x * blockDim.x + threadIdx.x;
    if (gid >= N_NODES * cols) return;
    int c = gid % cols;
    const float inv_n = 1.0f / (float)N_NODES;
    float m   = sums[c] * inv_n;
    float var = sums[cols + c] * inv_n - m * m;
    float y = (Z[gid] - m) * rsqrtf(var + 1e-5f) * g[c] + b[c];
    if (do_relu) y = fmaxf(y, 0.f);
    out[gid] = y;
}

// ---------------------------------------------------------------------------
// Pooling: pooled[batch[n],:] += h[n,:]; cnt[batch[n]] += 1; out = pooled/(cnt+1e-9)
// ---------------------------------------------------------------------------
__global__ void pool_accum(const float* __restrict__ h, const int* __restrict__ batch,
                           float* __restrict__ pooled, float* __restrict__ cnt) {
    int gid = blockIdx.x * blockDim.x + threadIdx.x;
    if (gid >= N_NODES * DIM) return;
    int n = gid >> 6;
    int d = gid & 63;
    int g = batch[n];
    atomicAdd(&pooled[(size_t)g * DIM + d], h[gid]);
    if (d == 0) atomicAdd(&cnt[g], 1.0f);
}

__global__ void pool_div(const float* __restrict__ pooled, const float* __restrict__ cnt,
                         float* __restrict__ out) {
    int gid = blockIdx.x * blockDim.x + threadIdx.x;
    if (gid >= N_GRAPHS * DIM) return;
    out[gid] = pooled[gid] / (cnt[gid >> 6] + 1e-9f);
}

// ---------------------------------------------------------------------------
// Host launcher
// ---------------------------------------------------------------------------
extern "C" void kernel_launch(void* const* d_in, const int* in_sizes, int n_in,
                              void* d_out, int out_size, void* d_ws, size_t ws_size,
                              hipStream_t stream) {
    // Inputs (setup_inputs order)
    const int*   x          = (const int*)  d_in[0];   // [N,9]
    const int*   edge_attr  = (const int*)  d_in[1];   // [E,3]
    const int*   edge_index = (const int*)  d_in[2];   // [2,E]
    const int*   batch      = (const int*)  d_in[3];   // [N]
    const float* atom_emb   = (const float*)d_in[4];   // [9,100,64]
    const float* bond_emb   = (const float*)d_in[5];   // [4,3,8,64]
    const float* W1         = (const float*)d_in[6];   // [4,64,128]
    const float* b1         = (const float*)d_in[7];   // [4,128]
    const float* g1         = (const float*)d_in[8];   // [4,128]
    const float* be1        = (const float*)d_in[9];   // [4,128]
    const float* W2         = (const float*)d_in[10];  // [4,128,64]
    const float* b2         = (const float*)d_in[11];  // [4,64]
    const float* eps        = (const float*)d_in[12];  // [4,1]
    const float* bn_g       = (const float*)d_in[13];  // [4,64]
    const float* bn_b       = (const float*)d_in[14];  // [4,64]

    // Workspace layout (floats)
    float* ws     = (float*)d_ws;
    float* h      = ws;                                  // N*64
    float* agg    = h      + (size_t)N_NODES * DIM;      // N*64
    float* z1     = agg    + (size_t)N_NODES * DIM;      // N*128
    float* stats  = z1     + (size_t)N_NODES * DIM2;     // 256
    float* pooled = stats  + 256;                        // 1024*64
    float* cnt    = pooled + (size_t)N_GRAPHS * DIM;     // 1024
    (void)ws_size; (void)n_in; (void)in_sizes; (void)out_size;

    const int BT = 256;
    dim3 blk(BT);

    // --- Atom encoder ---
    atom_encode<<<dim3((N_NODES * 16 + BT - 1) / BT), blk, 0, stream>>>(x, atom_emb, h);

    for (int l = 0; l < N_LAYERS; ++l) {
        // agg = 0
        zero_f<<<dim3((N_NODES * DIM + BT - 1) / BT), blk, 0, stream>>>(agg, N_NODES * DIM);

        // edge messages + scatter add
        edge_scatter<<<dim3((N_EDGES * 16 + BT - 1) / BT), blk, 0, stream>>>(
            edge_index, edge_attr, bond_emb + (size_t)l * 3 * 8 * DIM, h, agg);

        // z1 = ((1+eps)*h + agg) @ W1[l] + b1[l]    (WMMA f32)
        gemm1_wmma<<<dim3(N_NODES / 16, DIM2 / 16), dim3(32), 0, stream>>>(
            h, agg, eps, l, W1 + (size_t)l * DIM * DIM2, b1 + (size_t)l * DIM2, z1);

        // batchnorm(z1) + relu  -> z1 (in place)
        zero_f<<<dim3(1), blk, 0, stream>>>(stats, 2 * DIM2);
        bn_stats<<<dim3(512), blk, 0, stream>>>(z1, DIM2, stats);
        bn_apply<<<dim3((N_NODES * DIM2 + BT - 1) / BT), blk, 0, stream>>>(
            z1, z1, stats, g1 + (size_t)l * DIM2, be1 + (size_t)l * DIM2, DIM2, 1);

        // h = z1 @ W2[l] + b2[l]    (WMMA f32, written into h)
        gemm2_wmma<<<dim3(N_NODES / 16, DIM / 16), dim3(32), 0, stream>>>(
            z1, W2 + (size_t)l * DIM2 * DIM, b2 + (size_t)l * DIM, h);

        // batchnorm(h) (+ relu except last layer) -> h (in place)
        zero_f<<<dim3(1), blk, 0, stream>>>(stats, 2 * DIM);
        bn_stats<<<dim3(512), blk, 0, stream>>>(h, DIM, stats);
        bn_apply<<<dim3((N_NODES * DIM + BT - 1) / BT), blk, 0, stream>>>(
            h, h, stats, bn_g + (size_t)l * DIM, bn_b + (size_t)l * DIM, DIM,
            (l != N_LAYERS - 1) ? 1 : 0);
    }

    // --- Mean pooling per graph ---
    zero_f<<<dim3((N_GRAPHS * DIM + BT - 1) / BT), blk, 0, stream>>>(pooled, N_GRAPHS * DIM);
    zero_f<<<dim3(1), blk, 0, stream>>>(cnt, N_GRAPHS);
    pool_accum<<<dim3((N_NODES * DIM + BT - 1) / BT), blk, 0, stream>>>(h, batch, pooled, cnt);
    pool_div<<<dim3((N_GRAPHS * DIM + BT - 1) / BT), blk, 0, stream>>>(pooled, cnt, (float*)d_out);
}